// SSIMMetric_68006512164968
// MI455X (gfx1250) — compile-verified
//
#include <hip/hip_runtime.h>

typedef float v2f __attribute__((ext_vector_type(2)));
typedef float v8f __attribute__((ext_vector_type(8)));

#define SDIM   112
#define SS2    (SDIM * SDIM)          // 12544
#define SS3    (SDIM * SDIM * SDIM)   // 1404928
#define NBATCH 4
#define NSIG   5
#define KSTEPS 7                      // K = 28 >= 26 needed taps
#define WPB    8                      // waves per block (wave32)
#define BLOCK  256

// ---------------- Gaussian window helpers (computed in-register) -----------

__device__ __forceinline__ float gw_raw(int j) {
  float d = (float)(j - 5);
  return __expf(d * d * (-1.0f / 4.5f));   // sigma = 1.5 -> 2*sigma^2 = 4.5
}

__device__ __forceinline__ float inv_gauss_sum() {
  float s = 0.f;
#pragma unroll
  for (int j = 0; j <= 10; ++j) s += gw_raw(j);
  return 1.0f / s;
}

// Banded conv matrix: B[k][n] = g[k-n] if 0 <= k-n <= 10 else 0
__device__ __forceinline__ float bweight(int k, int n, float invs) {
  int j = k - n;
  float w = gw_raw(j) * invs;
  return (j >= 0 && j <= 10) ? w : 0.0f;
}

// ---------------- WMMA f32 16x16x4 wrapper ---------------------------------

__device__ __forceinline__ v8f wmma_f32_16x16x4(v2f a, v2f b, v8f c) {
  // 8-arg pattern per probe: (neg_a, A, neg_b, B, c_mod, C, reuse_a, reuse_b)
  return __builtin_amdgcn_wmma_f32_16x16x4_f32(false, a, false, b, (short)0, c,
                                               false, false);
}

// ---------------- Pass A: conv along W (x), build 5 signals ----------------
// One wave per tile: 16 lines (consecutive y, fixed z) x 16 output x.
__global__ __launch_bounds__(BLOCK) void ssim_passA(
    const float* __restrict__ P, const float* __restrict__ T,
    float* __restrict__ outA, int b, int zInLo, int planes, int sigStride) {
  __shared__ float lds[WPB][2][16 * 33];
  const int wave = threadIdx.x >> 5;
  const int lane = threadIdx.x & 31;
  const int w = blockIdx.x * WPB + wave;
  const int totalWaves = planes * 7 * 7;
  if (w >= totalWaves) return;   // uniform per wave

  const int tileX = w % 7;
  const int lg    = w / 7;          // line group (16 consecutive y lines)
  const int zi    = lg / 7;         // slab-local z plane
  const int yBase = (lg % 7) * 16;
  const int x0    = tileX * 16;
  const int q     = lane & 15;      // A row / B col / D col
  const int half  = lane >> 4;

  const float invs = inv_gauss_sum();
  float bf0[KSTEPS], bf1[KSTEPS];
#pragma unroll
  for (int kk = 0; kk < KSTEPS; ++kk) {
    bf0[kk] = bweight(kk * 4 + 2 * half + 0, q, invs);
    bf1[kk] = bweight(kk * 4 + 2 * half + 1, q, invs);
  }

  // Stage 16x32 tiles of p and t (x zero-padded), coalesced 128B rows.
  float* lp = lds[wave][0];
  float* lt = lds[wave][1];
  const int gx = x0 - 5 + lane;
  const bool xok = (gx >= 0) && (gx < SDIM);
  const size_t base =
      (size_t)b * SS3 + (size_t)(zInLo + zi) * SS2 + (size_t)yBase * SDIM;
#pragma unroll
  for (int i = 0; i < 16; ++i) {
    lp[i * 33 + lane] = xok ? P[base + (size_t)i * SDIM + gx] : 0.f;
    lt[i * 33 + lane] = xok ? T[base + (size_t)i * SDIM + gx] : 0.f;
  }
  __builtin_amdgcn_wave_barrier();

  v8f acc0 = {}, acc1 = {}, acc2 = {}, acc3 = {}, acc4 = {};
#pragma unroll
  for (int kk = 0; kk < KSTEPS; ++kk) {
    const int xi = kk * 4 + 2 * half;
    const float p0 = lp[q * 33 + xi], p1 = lp[q * 33 + xi + 1];
    const float t0 = lt[q * 33 + xi], t1 = lt[q * 33 + xi + 1];
    v2f bv; bv[0] = bf0[kk]; bv[1] = bf1[kk];
    v2f a;
    a[0] = p0;      a[1] = p1;      acc0 = wmma_f32_16x16x4(a, bv, acc0);
    a[0] = t0;      a[1] = t1;      acc1 = wmma_f32_16x16x4(a, bv, acc1);
    a[0] = p0 * p0; a[1] = p1 * p1; acc2 = wmma_f32_16x16x4(a, bv, acc2);
    a[0] = t0 * t0; a[1] = t1 * t1; acc3 = wmma_f32_16x16x4(a, bv, acc3);
    a[0] = p0 * t0; a[1] = p1 * t1; acc4 = wmma_f32_16x16x4(a, bv, acc4);
  }

  const size_t ob = (size_t)zi * SS2 + (size_t)yBase * SDIM + (size_t)x0 + q;
  v8f accs[NSIG] = {acc0, acc1, acc2, acc3, acc4};
#pragma unroll
  for (int s = 0; s < NSIG; ++s) {
#pragma unroll
    for (int v = 0; v < 8; ++v) {
      const int line = v + 8 * half;  // D row M -> line within group
      outA[(size_t)s * sigStride + ob + (size_t)line * SDIM] = accs[s][v];
    }
  }
}

// ---------------- Pass B: conv along H (y) ---------------------------------
// One wave per tile: 16 contiguous x (M) x 16 output y (N), fixed (sig,z).
__global__ __launch_bounds__(BLOCK) void ssim_passB(
    const float* __restrict__ inA, float* __restrict__ outB, int planes,
    int sigStride) {
  __shared__ float tr[WPB][16 * 17];
  const int wave = threadIdx.x >> 5;
  const int lane = threadIdx.x & 31;
  int w = blockIdx.x * WPB + wave;
  const int totalWaves = NSIG * planes * 7 * 7;
  if (w >= totalWaves) return;

  const int tileX = w % 7; w /= 7;
  const int tileY = w % 7; w /= 7;
  const int zi  = w % planes;
  const int sig = w / planes;
  const int q    = lane & 15;
  const int half = lane >> 4;

  const float invs = inv_gauss_sum();
  const size_t base = (size_t)sig * sigStride + (size_t)zi * SS2;
  const int xb = tileX * 16;
  const int y0 = tileY * 16;

  v8f acc = {};
#pragma unroll
  for (int kk = 0; kk < KSTEPS; ++kk) {
    v2f bv, a;
#pragma unroll
    for (int v = 0; v < 2; ++v) {
      const int k = kk * 4 + 2 * half + v;
      bv[v] = bweight(k, q, invs);
      const int yin = y0 - 5 + k;
      a[v] = (yin >= 0 && yin < SDIM)
                 ? inA[base + (size_t)yin * SDIM + xb + q]
                 : 0.f;
    }
    acc = wmma_f32_16x16x4(a, bv, acc);
  }

  // Transpose 16x16 through LDS so stores are contiguous along x.
  float* tt = tr[wave];
#pragma unroll
  for (int v = 0; v < 8; ++v) tt[q * 17 + (v + 8 * half)] = acc[v];
  __builtin_amdgcn_wave_barrier();
#pragma unroll
  for (int i = 0; i < 8; ++i) {
    const int row = 2 * i + half;  // output y offset
    outB[base + (size_t)(y0 + row) * SDIM + xb + q] = tt[row * 17 + q];
  }
}

// ---------------- Pass C: conv along D (z) + SSIM + reduction --------------
// Grid sized exactly: 98*ntz blocks, 8 waves each -> no inactive waves.
__global__ __launch_bounds__(BLOCK) void ssim_passC(
    const float* __restrict__ inB, float* __restrict__ partials,
    int partialBase, int ztBase, int zInLo, int ntz, int sigStride) {
  __shared__ float red[BLOCK];
  const int wave = threadIdx.x >> 5;
  const int lane = threadIdx.x & 31;
  int w = blockIdx.x * WPB + wave;

  const int tileX = w % 7; w /= 7;
  const int tz = w % ntz;
  const int y  = w / ntz;           // 0..111
  const int q    = lane & 15;
  const int half = lane >> 4;
  const int xb = tileX * 16;
  const int z0 = (ztBase + tz) * 16;

  const float invs = inv_gauss_sum();
  float bf0[KSTEPS], bf1[KSTEPS];
#pragma unroll
  for (int kk = 0; kk < KSTEPS; ++kk) {
    bf0[kk] = bweight(kk * 4 + 2 * half + 0, q, invs);
    bf1[kk] = bweight(kk * 4 + 2 * half + 1, q, invs);
  }

  v8f acc[NSIG] = {};
#pragma unroll
  for (int kk = 0; kk < KSTEPS; ++kk) {
    v2f bv; bv[0] = bf0[kk]; bv[1] = bf1[kk];
    const int k0 = kk * 4 + 2 * half;
#pragma unroll
    for (int s = 0; s < NSIG; ++s) {
      v2f a;
#pragma unroll
      for (int v = 0; v < 2; ++v) {
        const int zg = z0 - 5 + k0 + v;
        const bool ok = (zg >= 0) && (zg < SDIM);
        const size_t addr = (size_t)s * sigStride +
                            (size_t)(zg - zInLo) * SS2 + (size_t)y * SDIM +
                            xb + q;
        a[v] = ok ? inB[addr] : 0.f;
      }
      acc[s] = wmma_f32_16x16x4(a, bv, acc[s]);
    }
  }

  // SSIM map elements live per-lane, aligned across the 5 accumulators.
  float lsum = 0.f;
  const float c1 = 1e-4f, c2 = 9e-4f;
#pragma unroll
  for (int v = 0; v < 8; ++v) {
    const float mu1 = acc[0][v], mu2 = acc[1][v];
    const float mu1s = mu1 * mu1, mu2s = mu2 * mu2, mu12 = mu1 * mu2;
    const float s1  = acc[2][v] - mu1s;
    const float s2  = acc[3][v] - mu2s;
    const float s12 = acc[4][v] - mu12;
    const float num = (2.f * mu12 + c1) * (2.f * s12 + c2);
    const float den = (mu1s + mu2s + c1) * (s1 + s2 + c2);
    lsum += num / (den + 1e-8f);
  }

  // Deterministic block reduction (fixed order).
  red[threadIdx.x] = lsum;
  __syncthreads();
  if (threadIdx.x == 0) {
    float s = 0.f;
    for (int i = 0; i < BLOCK; ++i) s += red[i];
    partials[partialBase + blockIdx.x] = s;
  }
}

// ---------------- Final deterministic reduction ----------------------------
__global__ __launch_bounds__(256) void ssim_final(
    const float* __restrict__ partials, int n, float* __restrict__ out,
    float scale) {
  __shared__ float red[256];
  float s = 0.f;
  for (int i = threadIdx.x; i < n; i += 256) s += partials[i];
  red[threadIdx.x] = s;
  __syncthreads();
  if (threadIdx.x == 0) {
    float t = 0.f;
    for (int i = 0; i < 256; ++i) t += red[i];
    out[0] = t * scale;
  }
}

// ---------------- Host launch ----------------------------------------------
extern "C" void kernel_launch(void* const* d_in, const int* in_sizes, int n_in,
                              void* d_out, int out_size, void* d_ws,
                              size_t ws_size, hipStream_t stream) {
  (void)in_sizes; (void)n_in; (void)out_size;
  const float* P = (const float*)d_in[0];
  const float* T = (const float*)d_in[1];
  float* out = (float*)d_out;
  float* ws = (float*)d_ws;

  // Pick largest z-slab (in 16-out-tile units) fitting the workspace:
  // need 10 * planes * 12544 floats + partials. SZT=7 -> ~56.2 MB.
  int SZT = 7;
  while (SZT > 1) {
    int pmax = 16 * SZT + 10; if (pmax > SDIM) pmax = SDIM;
    size_t need = (size_t)10 * pmax * SS2 * sizeof(float) + 4096;
    if (need <= ws_size) break;
    --SZT;
  }
  int pmax = 16 * SZT + 10; if (pmax > SDIM) pmax = SDIM;
  const int sigStride = pmax * SS2;
  float* wsA = ws;                                  // 5 slab volumes (x-conv)
  float* wsB = ws + (size_t)NSIG * sigStride;       // 5 slab volumes (xy-conv)
  float* partials = ws + (size_t)2 * NSIG * sigStride;

  int partialBase = 0;
  for (int b = 0; b < NBATCH; ++b) {
    for (int zt0 = 0; zt0 < 7; zt0 += SZT) {
      const int ntz = (7 - zt0 < SZT) ? (7 - zt0) : SZT;
      const int zInLo = (zt0 * 16 - 5 > 0) ? (zt0 * 16 - 5) : 0;
      int zInHi = (zt0 + ntz) * 16 + 5; if (zInHi > SDIM) zInHi = SDIM;
      const int planes = zInHi - zInLo;

      const int wavesA = planes * 7 * 7;
      ssim_passA<<<(wavesA + WPB - 1) / WPB, BLOCK, 0, stream>>>(
          P, T, wsA, b, zInLo, planes, sigStride);

      const int wavesB = NSIG * planes * 7 * 7;
      ssim_passB<<<(wavesB + WPB - 1) / WPB, BLOCK, 0, stream>>>(
          wsA, wsB, planes, sigStride);

      const int blocksC = 98 * ntz;  // exactly 112*ntz*7 waves
      ssim_passC<<<blocksC, BLOCK, 0, stream>>>(wsB, partials, partialBase,
                                                zt0, zInLo, ntz, sigStride);
      partialBase += blocksC;
    }
  }
  ssim_final<<<1, 256, 0, stream>>>(partials, partialBase, out,
                                    1.0f / 5619712.0f);  // mean over 4*112^3
}